// SparseGCNConv_58411555225955
// MI455X (gfx1250) — compile-verified
//
#include <hip/hip_runtime.h>

typedef float v2f __attribute__((ext_vector_type(2)));
typedef float v8f __attribute__((ext_vector_type(8)));

#define NN 50000
#define DD 128
// M tiles: 50000/16 = 3125 exact. Edges: 1600000 / 8 per block = 200000 exact.

// Guaranteed-native FP32 scatter add: non-returning global_atomic_add_f32 at
// device scope (resolves in L2; multiple WGPs update the same agg rows).
// Tracked on STOREcnt; s_endpgm's implicit wait-idle covers completion.
__device__ __forceinline__ void atomic_add_f32_dev(float* p, float v) {
    asm volatile("global_atomic_add_f32 %0, %1, off scope:SCOPE_DEV"
                 :: "v"(p), "v"(v)
                 : "memory");
}

// ---------------- kernel 1: zero the aggregation workspace -----------------
__global__ void zero_ws(float4* __restrict__ ws, int n4) {
    int i = blockIdx.x * blockDim.x + threadIdx.x;
    int stride = gridDim.x * blockDim.x;
    float4 z = make_float4(0.f, 0.f, 0.f, 0.f);
    for (int j = i; j < n4; j += stride) ws[j] = z;
}

// ---------------- kernel 2: COO SpMM scatter (wave per edge) ---------------
__global__ void spmm_scatter(const int* __restrict__ row,
                             const int* __restrict__ col,
                             const float* __restrict__ vals,
                             const float* __restrict__ feat,
                             float* __restrict__ agg, int E) {
    unsigned gtid = blockIdx.x * blockDim.x + threadIdx.x;
    int e    = (int)(gtid >> 5);       // one wave32 per edge
    int lane = (int)(threadIdx.x & 31);
    if (e >= E) return;
    int   r = row[e];                  // uniform per wave (broadcast load)
    int   c = col[e];
    float v = vals[e];
    // 32 lanes x float4 = 128 dims
    float4 x = ((const float4*)(feat + (size_t)c * DD))[lane];
    float* dst = agg + (size_t)r * DD + lane * 4;
    atomic_add_f32_dev(dst + 0, v * x.x);
    atomic_add_f32_dev(dst + 1, v * x.y);
    atomic_add_f32_dev(dst + 2, v * x.z);
    atomic_add_f32_dev(dst + 3, v * x.w);
}

// ---------------- kernel 3: agg @ W + bias via FP32 WMMA -------------------
// Block = 256 threads = 8 waves. Block handles a 16x128 output slab
// (one M-tile of 16 rows); wave w handles N-columns [16w, 16w+16).
__global__ void __launch_bounds__(256) gemm_wmma(const float* __restrict__ agg,
                                                 const float* __restrict__ W,   // [128,128] row-major (K x N)
                                                 const float* __restrict__ bias,
                                                 float* __restrict__ out) {
    __shared__ float As[16][DD + 4];   // +4 pad: row stride 132 floats (528B, 16B aligned)

    int tid   = threadIdx.x;
    int tileM = blockIdx.x;            // 0..3124

    // Stage the 16x128 A tile: 512 float4, 2 per thread.
    const float4* src = (const float4*)(agg + (size_t)tileM * 16 * DD);
    #pragma unroll
    for (int it = tid; it < 512; it += 256) {
        int rrow = it >> 5;            // 32 float4 per row
        int ccol = it & 31;
        *((float4*)&As[rrow][ccol * 4]) = src[it];
    }
    __syncthreads();

    int wave = tid >> 5;               // N-tile index 0..7
    int lane = tid & 31;
    int half = lane >> 4;              // 0: K,K+1   1: K+2,K+3
    int mrow = lane & 15;
    int n    = wave * 16 + mrow;       // output column

    v8f acc = {};
    #pragma unroll
    for (int k = 0; k < DD; k += 4) {
        int kk = k + 2 * half;
        v2f a, b;
        // 32-bit A 16x4 layout: lanes 0-15 hold K=k,k+1; lanes 16-31 hold K=k+2,k+3
        a.x = As[mrow][kk];
        a.y = As[mrow][kk + 1];
        // 32-bit B 4x16 layout (transpose of A): same K split across lane halves
        b.x = W[(size_t)kk * DD + n];
        b.y = W[(size_t)(kk + 1) * DD + n];
        acc = __builtin_amdgcn_wmma_f32_16x16x4_f32(
                  /*neg_a=*/false, a, /*neg_b=*/false, b,
                  /*c_mod=*/(short)0, acc, /*reuse_a=*/false, /*reuse_b=*/false);
    }

    float bv = bias[n];
    size_t base = (size_t)tileM * 16 * DD + n;
    #pragma unroll
    for (int r = 0; r < 8; ++r) {
        // C/D layout: VGPR r -> M = r + 8*half, N = lane&15 (within tile)
        out[base + (size_t)(r + 8 * half) * DD] = acc[r] + bv;
    }
}

// ---------------------------------------------------------------------------
extern "C" void kernel_launch(void* const* d_in, const int* in_sizes, int n_in,
                              void* d_out, int out_size, void* d_ws, size_t ws_size,
                              hipStream_t stream) {
    const int*   row  = (const int*)d_in[0];
    const int*   col  = (const int*)d_in[1];
    const float* vals = (const float*)d_in[2];
    const float* feat = (const float*)d_in[3];
    const float* W    = (const float*)d_in[4];
    const float* bias = (const float*)d_in[5];
    float* out = (float*)d_out;
    float* agg = (float*)d_ws;         // 50000*128*4 = 25.6 MB scratch

    int E  = in_sizes[0];
    int n4 = NN * DD / 4;

    zero_ws<<<2048, 256, 0, stream>>>((float4*)agg, n4);

    int spmm_blocks = (int)(((long)E * 32 + 255) / 256);   // wave per edge
    spmm_scatter<<<spmm_blocks, 256, 0, stream>>>(row, col, vals, feat, agg, E);

    gemm_wmma<<<NN / 16, 256, 0, stream>>>(agg, W, bias, out);
}